// SSIMLoss_28750511079771
// MI455X (gfx1250) — compile-verified
//
#include <hip/hip_runtime.h>

typedef float v2f __attribute__((ext_vector_type(2)));
typedef float v8f __attribute__((ext_vector_type(8)));

#define PSTRIDE   28          // patch row stride (dwords): 26 cols + 2 pad, even for b64 align
#define PROWS_ALL 32          // allocate 32 rows (rows 26..31 zeroed) so chunk1 reads are defined
#define HT_STRIDE 36          // transposed-H row stride (dwords), even for b64 align
#define NTILES    49152       // 48 planes * 32*32 tiles
#define NPIX      12582912.0f // 16*3*512*512
#define C1V       1.0e-4f     // 0.01^2
#define C2V       9.0e-4f     // 0.03^2

// Normalized 11-tap Gaussian, sigma=1.5 (matches reference _gaussian_window)
__constant__ float G11[11] = {
    0.00102838f, 0.00759870f, 0.03600077f, 0.10936069f, 0.21300553f,
    0.26601172f, 0.21300553f, 0.10936069f, 0.03600077f, 0.00759870f,
    0.00102838f
};

__device__ __forceinline__ v8f wmma_f32_16x16x4(v2f a, v2f b, v8f c) {
    // (neg_a, A, neg_b, B, c_mod, C, reuse_a, reuse_b)
    return __builtin_amdgcn_wmma_f32_16x16x4_f32(false, a, false, b, (short)0, c,
                                                 false, false);
}

__global__ __launch_bounds__(32) void ssim_tile_kernel(
    const float* __restrict__ pred, const float* __restrict__ targ,
    float* __restrict__ partials)
{
    __shared__ float ldsP[2][PROWS_ALL * PSTRIDE];   // zero-padded 26x26 patches (p, t)
    __shared__ float ldsH[5][16 * HT_STRIDE];        // transposed horizontal results x5

    const int lane = threadIdx.x;    // 0..31, wave32
    const int h    = lane >> 4;      // lane-half
    const int m    = lane & 15;      // M (A) / N (B,C,D) index

    const int tile  = blockIdx.x;
    const int plane = tile >> 10;            // 1024 tiles per 512x512 plane
    const int trem  = tile & 1023;
    const int ty    = trem >> 5;
    const int tx    = trem & 31;
    const int y0    = ty * 16 - 5;           // patch origin (SAME pad = 5)
    const int x0    = tx * 16 - 5;

    const float* p = pred + (size_t)plane * 262144;
    const float* t = targ + (size_t)plane * 262144;

    // Banded Gaussian weight fragments. For reg r, lane-half h: K = 4k + r + 2h.
    // Horizontal B role: Wh[K][n] = g[K-n]; vertical A role: Wv[m][K] = g[K-m].
    // With n == m == lane&15 these are the same per-lane values.
    v2f wf[7];
    #pragma unroll
    for (int k = 0; k < 7; ++k) {
        const int d0 = 4 * k + 2 * h - m;
        const int d1 = d0 + 1;
        wf[k].x = (d0 >= 0 && d0 <= 10) ? G11[d0] : 0.0f;
        wf[k].y = (d1 >= 0 && d1 <= 10) ? G11[d1] : 0.0f;
    }

    // Cooperative zero-padded patch load (entire 32x28 buffer written => no garbage)
    for (int i = lane; i < PROWS_ALL * PSTRIDE; i += 32) {
        const int r  = i / PSTRIDE;
        const int c  = i - r * PSTRIDE;
        const int gy = y0 + r;
        const int gx = x0 + c;
        float vp = 0.0f, vt = 0.0f;
        if (r < 26 && c < 26 && gy >= 0 && gy < 512 && gx >= 0 && gx < 512) {
            const int off = gy * 512 + gx;
            vp = p[off];
            vt = t[off];
        }
        ldsP[0][i] = vp;
        ldsP[1][i] = vt;
    }
    __syncthreads();

    // ---------------- Horizontal pass: H = Patch(32x28) x Wh(28x16) ----------------
    #pragma unroll
    for (int rc = 0; rc < 2; ++rc) {
        v8f c0 = {}, c1 = {}, c2 = {}, c3 = {}, c4 = {};
        const int rowbase = (rc * 16 + m) * PSTRIDE + 2 * h;
        #pragma unroll
        for (int k = 0; k < 7; ++k) {
            const v2f ap  = *(const v2f*)&ldsP[0][rowbase + 4 * k]; // ds_load_b64
            const v2f at  = *(const v2f*)&ldsP[1][rowbase + 4 * k];
            const v2f app = ap * ap;   // fragment-wise products == gathered p^2 etc.
            const v2f att = at * at;
            const v2f apt = ap * at;
            c0 = wmma_f32_16x16x4(ap,  wf[k], c0);
            c1 = wmma_f32_16x16x4(at,  wf[k], c1);
            c2 = wmma_f32_16x16x4(app, wf[k], c2);
            c3 = wmma_f32_16x16x4(att, wf[k], c3);
            c4 = wmma_f32_16x16x4(apt, wf[k], c4);
        }
        // Store C/D frags transposed: ldsH[q][n * HT_STRIDE + rowIdx]
        // C/D layout: VGPR v holds row (rc*16 + v + 8*h), column = lane&15.
        const int colbase = m * HT_STRIDE + rc * 16 + 8 * h;
        #pragma unroll
        for (int v = 0; v < 8; ++v) {
            ldsH[0][colbase + v] = c0[v];
            ldsH[1][colbase + v] = c1[v];
            ldsH[2][colbase + v] = c2[v];
            ldsH[3][colbase + v] = c3[v];
            ldsH[4][colbase + v] = c4[v];
        }
    }
    __syncthreads();

    // ---------------- Vertical pass: O = Wv(16x28) x H(28x16) ----------------
    v8f o0 = {}, o1 = {}, o2 = {}, o3 = {}, o4 = {};
    #pragma unroll
    for (int k = 0; k < 7; ++k) {
        const int off = m * HT_STRIDE + 4 * k + 2 * h;   // {H[K][n], H[K+1][n]}
        const v2f b0 = *(const v2f*)&ldsH[0][off];
        const v2f b1 = *(const v2f*)&ldsH[1][off];
        const v2f b2 = *(const v2f*)&ldsH[2][off];
        const v2f b3 = *(const v2f*)&ldsH[3][off];
        const v2f b4 = *(const v2f*)&ldsH[4][off];
        o0 = wmma_f32_16x16x4(wf[k], b0, o0);
        o1 = wmma_f32_16x16x4(wf[k], b1, o1);
        o2 = wmma_f32_16x16x4(wf[k], b2, o2);
        o3 = wmma_f32_16x16x4(wf[k], b3, o3);
        o4 = wmma_f32_16x16x4(wf[k], b4, o4);
    }

    // ---------------- SSIM map + tile partial sum ----------------
    float partial = 0.0f;
    #pragma unroll
    for (int v = 0; v < 8; ++v) {
        const float mu1  = o0[v];
        const float mu2  = o1[v];
        const float m1s  = mu1 * mu1;
        const float m2s  = mu2 * mu2;
        const float m12  = mu1 * mu2;
        const float s1   = o2[v] - m1s;
        const float s2   = o3[v] - m2s;
        const float s12  = o4[v] - m12;
        const float num  = (2.0f * m12 + C1V) * (2.0f * s12 + C2V);
        const float den  = (m1s + m2s + C1V) * (s1 + s2 + C2V);
        partial += num / den;
    }
    #pragma unroll
    for (int off = 16; off > 0; off >>= 1)
        partial += __shfl_xor(partial, off, 32);
    if (lane == 0)
        partials[blockIdx.x] = partial;   // fully overwritten every launch
}

__global__ __launch_bounds__(256) void ssim_reduce_kernel(
    const float* __restrict__ partials, float* __restrict__ out)
{
    __shared__ float sbuf[256];
    float s = 0.0f;
    for (int i = threadIdx.x; i < NTILES; i += 256)
        s += partials[i];                 // fixed order -> deterministic
    sbuf[threadIdx.x] = s;
    __syncthreads();
    for (int o = 128; o > 0; o >>= 1) {
        if (threadIdx.x < o) sbuf[threadIdx.x] += sbuf[threadIdx.x + o];
        __syncthreads();
    }
    if (threadIdx.x == 0)
        out[0] = 1.0f - sbuf[0] / NPIX;
}

extern "C" void kernel_launch(void* const* d_in, const int* in_sizes, int n_in,
                              void* d_out, int out_size, void* d_ws, size_t ws_size,
                              hipStream_t stream) {
    (void)in_sizes; (void)n_in; (void)out_size; (void)ws_size;
    const float* pred = (const float*)d_in[0];
    const float* targ = (const float*)d_in[1];
    float* out       = (float*)d_out;
    float* partials  = (float*)d_ws;     // NTILES * 4 bytes

    ssim_tile_kernel<<<NTILES, 32, 0, stream>>>(pred, targ, partials);
    ssim_reduce_kernel<<<1, 256, 0, stream>>>(partials, out);
}